// PatchID_25623774888148
// MI455X (gfx1250) — compile-verified
//
#include <hip/hip_runtime.h>
#include <math.h>

// ---------------- problem constants ----------------
#define NND   883
#define BATCH 8
#define TOKR  (BATCH*NND)     // 7064 real tokens
#define TOK   7104            // padded to multiple of 64 (111 blocks of 4 M-tiles)
#define DIMS  160
#define HID   320
#define RP    8               // R_PATCH
#define ILEN  12
#define OLEN  12
#define PRED_SIZE (BATCH*OLEN*NND)   // 84768

typedef float v2f __attribute__((ext_vector_type(2)));
typedef float v8f __attribute__((ext_vector_type(8)));

__device__ __forceinline__ v8f wmma_f32(v2f a, v2f b, v8f c) {
    return __builtin_amdgcn_wmma_f32_16x16x4_f32(false, a, false, b,
                                                 (short)0, c, false, false);
}

// ---------------------------------------------------------------------------
// 1) Patch assignment
// ---------------------------------------------------------------------------
__global__ void assign_kernel(const float* __restrict__ node_table,
                              const float* __restrict__ patch_emb,
                              float* __restrict__ probs_st,   // (883,8) -> d_out tail
                              float* __restrict__ node_soft,  // (883,32)
                              int* __restrict__ patch_ids,    // (883)
                              int* __restrict__ pos,          // (883)
                              int* __restrict__ patch_count,  // (8)
                              int* __restrict__ members)      // (8,883)
{
    __shared__ int s_pid[NND];
    __shared__ int s_cnt[RP];
    int t = threadIdx.x;
    if (t < RP) s_cnt[t] = 0;
    __syncthreads();

    for (int n = t; n < NND; n += blockDim.x) {
        float lg[RP];
        for (int r = 0; r < RP; ++r) {
            float s = 0.f;
            const float* nt = node_table + n*32;
            const float* pe = patch_emb + r*32;
            for (int k = 0; k < 32; ++k) s += nt[k]*pe[k];
            lg[r] = s;
        }
        float mx = lg[0]; int am = 0;
        for (int r = 1; r < RP; ++r) if (lg[r] > mx) { mx = lg[r]; am = r; }
        float pr[RP], se = 0.f;
        for (int r = 0; r < RP; ++r) { pr[r] = expf(lg[r]-mx); se += pr[r]; }
        for (int r = 0; r < RP; ++r) pr[r] /= se;
        s_pid[n] = am;
        patch_ids[n] = am;
        for (int r = 0; r < RP; ++r) {
            float hard = (r == am) ? 1.f : 0.f;
            probs_st[n*RP + r] = (hard - pr[r]) + pr[r];
        }
        for (int k = 0; k < 32; ++k) {
            float s = 0.f;
            for (int r = 0; r < RP; ++r) s += pr[r]*patch_emb[r*32 + k];
            node_soft[n*32 + k] = s;
        }
    }
    __syncthreads();
    for (int n = t; n < NND; n += blockDim.x) {
        int myp = s_pid[n];
        int c = 0;
        for (int m = 0; m < n; ++m) if (s_pid[m] == myp) ++c;
        pos[n] = c;
        members[myp*NND + c] = n;
        atomicAdd(&s_cnt[myp], 1);
    }
    __syncthreads();
    if (t < RP) patch_count[t] = s_cnt[t];
}

// ---------------------------------------------------------------------------
// 2) Embedding (pad rows [TOKR,TOK) zeroed)
// ---------------------------------------------------------------------------
__global__ void embed_kernel(const float* __restrict__ x,
                             const int*   __restrict__ te,
                             const float* __restrict__ conv_W,  // (64,3,12)
                             const float* __restrict__ conv_b,
                             const float* __restrict__ tod_emb, // (288,32)
                             const float* __restrict__ dow_emb, // (7,32)
                             const float* __restrict__ node_soft,
                             float* __restrict__ X0)            // (TOK,160)
{
    int tok = blockIdx.x;
    int d = threadIdx.x;            // blockDim = 160
    if (tok >= TOKR) { X0[(size_t)tok*DIMS + d] = 0.f; return; }
    int b = tok / NND, n = tok % NND;
    float out;
    if (d < 64) {
        float s = conv_b[d];
        const float* cw = conv_W + d*36;
        for (int l = 0; l < ILEN; ++l) {
            size_t base = (size_t)(b*ILEN + l)*NND + n;
            float x0 = x[base];
            float t0 = (float)te[base*2 + 0] * (1.f/288.f);
            float t1 = (float)te[base*2 + 1] * (1.f/7.f);
            s += x0*cw[l] + t0*cw[12 + l] + t1*cw[24 + l];
        }
        out = s;
    } else if (d < 96) {
        int tid_ = te[((size_t)(b*ILEN + 11)*NND + n)*2 + 0];
        out = tod_emb[tid_*32 + (d - 64)];
    } else if (d < 128) {
        int diw = te[((size_t)(b*ILEN + 11)*NND + n)*2 + 1];
        out = dow_emb[diw*32 + (d - 96)];
    } else {
        out = node_soft[n*32 + (d - 128)];
    }
    X0[(size_t)tok*DIMS + d] = out;
}

// ---------------------------------------------------------------------------
// 3) WMMA f32 GEMM, 4 M-tiles (64 rows) x 16 cols per wave.
//    K,N compile-time so all inner-loop loads use immediate offsets.
//    flags: bit0 = exact GELU, bit1 = add resid (Y may alias resid).
// ---------------------------------------------------------------------------
template<int K, int N>
__global__ void gemm_wmma(const float* __restrict__ X,
                          const float* W, const float* bias,
                          float* Y, const float* resid,
                          int M, int flags,
                          int wz_stride, int yz_stride, int bz_stride)
{
    int z = blockIdx.z;
    W += (size_t)z * wz_stride;
    Y += (size_t)z * yz_stride;
    if (bias) bias += (size_t)z * bz_stride;

    const int lane = threadIdx.x & 31;
    const int half = lane >> 4, lr = lane & 15;
    const int m0 = blockIdx.x * 64;
    const int n0 = blockIdx.y * 16;

    // A fragment row pointers (8-byte aligned: K even, 2*half even)
    const float* a0 = X + (size_t)(m0 + lr) * K + 2*half;
    const float* a1 = a0 + 16*K;
    const float* a2 = a1 + 16*K;
    const float* a3 = a2 + 16*K;
    // B fragment pointer: rows 2*half and 2*half+1 of the K-panel, column n0+lr
    const float* wp = W + (size_t)(2*half) * N + n0 + lr;

    v8f acc[4] = {v8f{}, v8f{}, v8f{}, v8f{}};
#pragma unroll 8
    for (int k0 = 0; k0 < K; k0 += 4) {
        v2f b;
        b.x = wp[0];
        b.y = wp[N];
        acc[0] = wmma_f32(*(const v2f*)a0, b, acc[0]);
        acc[1] = wmma_f32(*(const v2f*)a1, b, acc[1]);
        acc[2] = wmma_f32(*(const v2f*)a2, b, acc[2]);
        acc[3] = wmma_f32(*(const v2f*)a3, b, acc[3]);
        a0 += 4; a1 += 4; a2 += 4; a3 += 4;
        wp += 4*N;
    }

    float bn = bias ? bias[n0 + lr] : 0.f;
#pragma unroll
    for (int mt = 0; mt < 4; ++mt) {
#pragma unroll
        for (int j = 0; j < 8; ++j) {
            int row = m0 + mt*16 + j + 8*half;
            if (row >= M) continue;
            float v = acc[mt][j] + bn;
            if (flags & 1) v = 0.5f * v * (1.f + erff(v * 0.70710678118654752f));
            size_t idx = (size_t)row * N + n0 + lr;
            if (flags & 2) v += resid[idx];
            Y[idx] = v;
        }
    }
}

// ---------------------------------------------------------------------------
// 4) LayerNorm per token (160 threads/block).
// ---------------------------------------------------------------------------
__global__ void ln_kernel(const float* __restrict__ X, float* __restrict__ Y,
                          const float* __restrict__ g, const float* __restrict__ b)
{
    __shared__ float red[DIMS];
    __shared__ float stat[2];
    int tok = blockIdx.x, d = threadIdx.x;
    const float* x = X + (size_t)tok*DIMS;
    float v = x[d];
    red[d] = v; __syncthreads();
    for (int s = 80; s >= 5; s >>= 1) { if (d < s) red[d] += red[d + s]; __syncthreads(); }
    if (d == 0) stat[0] = (red[0]+red[1]+red[2]+red[3]+red[4]) * (1.f/DIMS);
    __syncthreads();
    float mu = stat[0];
    float df = v - mu;
    red[d] = df*df; __syncthreads();
    for (int s = 80; s >= 5; s >>= 1) { if (d < s) red[d] += red[d + s]; __syncthreads(); }
    if (d == 0) {
        float var = (red[0]+red[1]+red[2]+red[3]+red[4]) * (1.f/DIMS);
        stat[1] = 1.f / sqrtf(var + 1e-5f);
    }
    __syncthreads();
    Y[(size_t)tok*DIMS + d] = df * stat[1] * g[d] + b[d];
}

// ---------------------------------------------------------------------------
// 5) Grouped attention. mode 0: intra-patch; mode 1: same-rank (<=8 members).
//    q staged in LDS; weights normalized before the weighted sum (matches w@v).
// ---------------------------------------------------------------------------
#define MAXG 896
__global__ void attn_kernel(const float* __restrict__ Q,
                            const float* __restrict__ Kk,
                            const float* __restrict__ V,
                            float* __restrict__ O,
                            const int* __restrict__ patch_ids,
                            const int* __restrict__ pos,
                            const int* __restrict__ patch_count,
                            const int* __restrict__ members,
                            int mode)
{
    int tok = blockIdx.x;
    int t = threadIdx.x;                 // blockDim = 256
    if (tok >= TOKR) {                   // pad rows -> zeros
        for (int d = t; d < DIMS; d += 256) O[(size_t)tok*DIMS + d] = 0.f;
        return;
    }
    int b = tok / NND, n = tok % NND;
    __shared__ float sq[DIMS];
    __shared__ float sc[MAXG];
    __shared__ int   mem[MAXG];
    __shared__ int   scnt;
    __shared__ float red[256];

    for (int d = t; d < DIMS; d += 256) sq[d] = Q[(size_t)tok*DIMS + d];

    if (mode == 0) {
        int r = patch_ids[n];
        int cnt = patch_count[r];
        if (t == 0) scnt = cnt;
        for (int i = t; i < cnt; i += 256) mem[i] = members[r*NND + i];
    } else {
        if (t == 0) {
            int p = pos[n], c = 0;
            for (int r = 0; r < RP; ++r)
                if (patch_count[r] > p) mem[c++] = members[r*NND + p];
            scnt = c;
        }
    }
    __syncthreads();
    int cnt = scnt;
    for (int i = t; i < cnt; i += 256) {
        const float* kp = Kk + ((size_t)b*NND + mem[i])*DIMS;
        float s = 0.f;
        for (int d = 0; d < DIMS; ++d) s += sq[d]*kp[d];
        sc[i] = s * 0.07905694150420949f;   // 1/sqrt(160)
    }
    __syncthreads();
    float lm = -3.0e38f;
    for (int i = t; i < cnt; i += 256) lm = fmaxf(lm, sc[i]);
    red[t] = lm; __syncthreads();
    for (int s = 128; s > 0; s >>= 1) { if (t < s) red[t] = fmaxf(red[t], red[t+s]); __syncthreads(); }
    float mx = red[0]; __syncthreads();
    float ls = 0.f;
    for (int i = t; i < cnt; i += 256) { float e = expf(sc[i]-mx); sc[i] = e; ls += e; }
    red[t] = ls; __syncthreads();
    for (int s = 128; s > 0; s >>= 1) { if (t < s) red[t] += red[t+s]; __syncthreads(); }
    float inv = 1.f / red[0];
    for (int i = t; i < cnt; i += 256) sc[i] *= inv;
    __syncthreads();
    for (int d = t; d < DIMS; d += 256) {
        float o = 0.f;
        for (int i = 0; i < cnt; ++i)
            o += sc[i] * V[((size_t)b*NND + mem[i])*DIMS + d];
        O[(size_t)tok*DIMS + d] = o;
    }
}

// ---------------------------------------------------------------------------
// 6) Output head: pred[b,o,n,0] = pf[b,n,:] . out_W[o,:] + out_b[o]
// ---------------------------------------------------------------------------
__global__ void out_kernel(const float* __restrict__ pf,
                           const float* __restrict__ out_W,
                           const float* __restrict__ out_b,
                           float* __restrict__ pred)
{
    int idx = blockIdx.x * blockDim.x + threadIdx.x;
    if (idx >= TOKR*OLEN) return;
    int tok = idx / OLEN, o = idx % OLEN;
    int b = tok / NND, n = tok % NND;
    const float* p = pf + (size_t)tok*DIMS;
    const float* w = out_W + o*DIMS;
    float s = 0.f;
    for (int d = 0; d < DIMS; ++d) s += p[d]*w[d];
    pred[(size_t)(b*OLEN + o)*NND + n] = s + out_b[o];
}

// ---------------------------------------------------------------------------
extern "C" void kernel_launch(void* const* d_in, const int* in_sizes, int n_in,
                              void* d_out, int out_size, void* d_ws, size_t ws_size,
                              hipStream_t stream)
{
    const float* x        = (const float*)d_in[0];
    const int*   te       = (const int*)  d_in[1];
    const float* patch_emb= (const float*)d_in[2];
    const float* node_tab = (const float*)d_in[3];
    const float* tod_emb  = (const float*)d_in[4];
    const float* dow_emb  = (const float*)d_in[5];
    const float* conv_W   = (const float*)d_in[6];
    const float* conv_b   = (const float*)d_in[7];
    const float* ln_scale = (const float*)d_in[8];
    const float* ln_bias  = (const float*)d_in[9];
    const float* attn_W   = (const float*)d_in[10];
    const float* attn_b   = (const float*)d_in[11];
    const float* mlp_W1   = (const float*)d_in[12];
    const float* mlp_b1   = (const float*)d_in[13];
    const float* mlp_W2   = (const float*)d_in[14];
    const float* mlp_b2   = (const float*)d_in[15];
    const float* out_W    = (const float*)d_in[16];
    const float* out_b    = (const float*)d_in[17];

    float* pred     = (float*)d_out;
    float* probs_st = pred + PRED_SIZE;

    // workspace carve-up (all fp32 / int32, 4B aligned)
    float* ws   = (float*)d_ws;
    float* pf   = ws;                       // TOK*160
    float* xn   = pf  + (size_t)TOK*DIMS;   // TOK*160
    float* qkv  = xn  + (size_t)TOK*DIMS;   // 3*TOK*160
    float* oat  = qkv + (size_t)3*TOK*DIMS; // TOK*160
    float* h1   = oat + (size_t)TOK*DIMS;   // TOK*320
    float* nsft = h1  + (size_t)TOK*HID;    // 883*32
    int*   ip          = (int*)(nsft + NND*32);
    int*   patch_ids   = ip;
    int*   pos         = ip + NND;
    int*   patch_count = ip + 2*NND;
    int*   members     = ip + 2*NND + RP;   // 8*883

    assign_kernel<<<1, 256, 0, stream>>>(node_tab, patch_emb, probs_st, nsft,
                                         patch_ids, pos, patch_count, members);
    embed_kernel<<<TOK, DIMS, 0, stream>>>(x, te, conv_W, conv_b, tod_emb,
                                           dow_emb, nsft, pf);

    const int MB = TOK/64;                  // 111 blocks of 4 M-tiles
    for (int l = 0; l < 3; ++l) {
        for (int blk = 0; blk < 2; ++blk) {
            const float* Wl = attn_W + (size_t)((l*2 + blk)*4) * DIMS*DIMS;
            const float* bl = attn_b + (size_t)((l*2 + blk)*4) * DIMS;
            const float* g0 = ln_scale + (size_t)(l*4 + blk*2)     * DIMS;
            const float* b0 = ln_bias  + (size_t)(l*4 + blk*2)     * DIMS;
            const float* g1 = ln_scale + (size_t)(l*4 + blk*2 + 1) * DIMS;
            const float* b1 = ln_bias  + (size_t)(l*4 + blk*2 + 1) * DIMS;
            const float* W1 = mlp_W1 + (size_t)(l*2 + blk) * DIMS*HID;
            const float* c1 = mlp_b1 + (size_t)(l*2 + blk) * HID;
            const float* W2 = mlp_W2 + (size_t)(l*2 + blk) * HID*DIMS;
            const float* c2 = mlp_b2 + (size_t)(l*2 + blk) * DIMS;

            // pre-attention LN
            ln_kernel<<<TOK, DIMS, 0, stream>>>(pf, xn, g0, b0);
            // Q,K,V in one z-batched launch
            gemm_wmma<DIMS, DIMS><<<dim3(MB, DIMS/16, 3), 32, 0, stream>>>(
                xn, Wl, bl, qkv, nullptr, TOK, 0,
                DIMS*DIMS, TOK*DIMS, DIMS);
            // grouped attention (blk 0 = intra-patch, blk 1 = same-rank)
            attn_kernel<<<TOK, 256, 0, stream>>>(
                qkv, qkv + (size_t)TOK*DIMS, qkv + (size_t)2*TOK*DIMS, oat,
                patch_ids, pos, patch_count, members, blk);
            // output projection + residual
            gemm_wmma<DIMS, DIMS><<<dim3(MB, DIMS/16, 1), 32, 0, stream>>>(
                oat, Wl + 3*DIMS*DIMS, bl + 3*DIMS, pf, pf,
                TOK, 2, 0, 0, 0);
            // pre-MLP LN
            ln_kernel<<<TOK, DIMS, 0, stream>>>(pf, xn, g1, b1);
            // MLP: gelu(xn@W1+b1) @ W2 + b2, residual
            gemm_wmma<DIMS, HID><<<dim3(MB, HID/16, 1), 32, 0, stream>>>(
                xn, W1, c1, h1, nullptr, TOK, 1, 0, 0, 0);
            gemm_wmma<HID, DIMS><<<dim3(MB, DIMS/16, 1), 32, 0, stream>>>(
                h1, W2, c2, pf, pf, TOK, 2, 0, 0, 0);
        }
    }

    out_kernel<<<(TOKR*OLEN + 255)/256, 256, 0, stream>>>(pf, out_W, out_b, pred);
}